// NSScan_40836549050610
// MI455X (gfx1250) — compile-verified
//
#include <hip/hip_runtime.h>
#include <cstdint>

// Problem geometry (fixed by the reference)
#define Hh    128
#define Ww    128
#define Cc    128
#define Ll    (Hh * Ww)       // 16384
#define Nn    8
#define NDIR  4
#define SHF   2

#define WAVES_PER_BLOCK 8
#define NSLOT           4
#define SLOT_BYTES      (Cc * 4)   // 512 B = one pixel's channel vector = 32 lanes * 16 B

// Boustrophedon index math (STRIPE=4 => "local%2" == parity of row/col).
// Given source pixel (r,c) in the rolled frame, position of that pixel in each
// direction's sequence:
__device__ __forceinline__ void seq_positions(int r, int c, int j[4]) {
    const int j0 = r * Ww + ((r & 1) ? (Ww - 1 - c) : c);   // h_fwd
    const int j2 = c * Hh + ((c & 1) ? (Hh - 1 - r) : r);   // v_fwd
    j[0] = j0;
    j[1] = (Ll - 1) - j0;                                    // h_bwd
    j[2] = j2;
    j[3] = (Ll - 1) - j2;                                    // v_bwd
}

// ---------------------------------------------------------------------------
// Kernel 1: scanned = gather of rolled x into 4 direction sequences.
// Source-major: read each 512B pixel of x ONCE (async -> LDS), scatter 4
// async stores from LDS. Pure ASYNCcnt pipeline, no VGPR data movement.
// One wave handles NSLOT pixels per launch.
// ---------------------------------------------------------------------------
__global__ __launch_bounds__(256) void nss_scatter_async(
    const float* __restrict__ x, float* __restrict__ scanned)
{
    __shared__ float4 sh[WAVES_PER_BLOCK * NSLOT * 32];     // 16 KB / block

    const int lane = threadIdx.x & 31;
    const int wave = threadIdx.x >> 5;

    // LDS byte address of this lane's 16B cell in slot 0 of this wave's slice.
    // (low 32 bits of the flat address of a __shared__ object == LDS offset)
    const uint32_t lds0 = (uint32_t)(uintptr_t)(&sh[0])
                        + (uint32_t)wave * (NSLOT * SLOT_BYTES)
                        + (uint32_t)lane * 16u;

    const int wid = blockIdx.x * WAVES_PER_BLOCK + wave;    // global wave id
    const int p0  = wid * NSLOT;                            // first pixel

    // ---- stage NSLOT pixels into LDS (async) ----
    #pragma unroll
    for (int s = 0; s < NSLOT; ++s) {
        const int p   = p0 + s;
        const int n   = p >> 14;                // p / (H*W)
        const int rem = p & (Ll - 1);
        const int r   = rem >> 7;
        const int c   = rem & (Ww - 1);
        // roll(-2,-2): rolled[r,c] = x[(r+2)%H, (c+2)%W]
        const int rs = (r + SHF) & (Hh - 1);
        const int cs = (c + SHF) & (Ww - 1);
        const float* src = x + ((size_t)((n * Ll) + rs * Ww + cs) << 7);
        const uint64_t g = (uint64_t)(uintptr_t)src + (uint64_t)(lane * 16);
        const uint32_t l = lds0 + (uint32_t)(s * SLOT_BYTES);
        asm volatile("global_load_async_to_lds_b128 %0, %1, off"
                     :: "v"(l), "v"(g) : "memory");
    }
    asm volatile("s_wait_asynccnt 0" ::: "memory");

    // ---- scatter each staged pixel to its 4 sequence positions (async) ----
    #pragma unroll
    for (int s = 0; s < NSLOT; ++s) {
        const int p   = p0 + s;
        const int n   = p >> 14;
        const int rem = p & (Ll - 1);
        const int r   = rem >> 7;
        const int c   = rem & (Ww - 1);
        int j[4];
        seq_positions(r, c, j);
        const uint32_t l = lds0 + (uint32_t)(s * SLOT_BYTES);
        #pragma unroll
        for (int d = 0; d < NDIR; ++d) {
            float* dst = scanned + ((size_t)(((d * Nn) + n) * Ll + j[d]) << 7);
            const uint64_t g = (uint64_t)(uintptr_t)dst + (uint64_t)(lane * 16);
            asm volatile("global_store_async_from_lds_b128 %0, %1, off"
                         :: "v"(g), "v"(l) : "memory");
        }
    }
    asm volatile("s_wait_asynccnt 0" ::: "memory");
}

// ---------------------------------------------------------------------------
// Kernel 2: rec = roll(+2,+2) of mean of 4 inverse-gathered chunks of x_seqs.
// Output-major: one wave per output pixel, lane = float4 of channels.
// 4 gathered 512B reads + 1 coalesced 512B write per wave.
// ---------------------------------------------------------------------------
__global__ __launch_bounds__(256) void nss_merge(
    const float* __restrict__ xseqs, float* __restrict__ rec)
{
    const int lane = threadIdx.x & 31;
    const int wave = threadIdx.x >> 5;
    const int pid  = blockIdx.x * WAVES_PER_BLOCK + wave;   // output pixel

    const int n   = pid >> 14;
    const int rem = pid & (Ll - 1);
    const int h   = rem >> 7;
    const int w   = rem & (Ww - 1);
    // roll(+2,+2): rec[h,w] = pre[(h-2)%H, (w-2)%W]
    const int r = (h - SHF) & (Hh - 1);
    const int c = (w - SHF) & (Ww - 1);

    int j[4];
    seq_positions(r, c, j);

    const float4* s0 = (const float4*)(xseqs + ((size_t)((0 * Nn + n) * Ll + j[0]) << 7)) + lane;
    const float4* s1 = (const float4*)(xseqs + ((size_t)((1 * Nn + n) * Ll + j[1]) << 7)) + lane;
    const float4* s2 = (const float4*)(xseqs + ((size_t)((2 * Nn + n) * Ll + j[2]) << 7)) + lane;
    const float4* s3 = (const float4*)(xseqs + ((size_t)((3 * Nn + n) * Ll + j[3]) << 7)) + lane;

    const float4 a = *s0, b = *s1, e = *s2, f = *s3;
    float4 o;
    // Match reference accumulation order: ((c0+c1)+c2)+c3, then /4 (exact *0.25)
    o.x = (((a.x + b.x) + e.x) + f.x) * 0.25f;
    o.y = (((a.y + b.y) + e.y) + f.y) * 0.25f;
    o.z = (((a.z + b.z) + e.z) + f.z) * 0.25f;
    o.w = (((a.w + b.w) + e.w) + f.w) * 0.25f;

    *((float4*)(rec + ((size_t)pid << 7)) + lane) = o;
}

extern "C" void kernel_launch(void* const* d_in, const int* in_sizes, int n_in,
                              void* d_out, int out_size, void* d_ws, size_t ws_size,
                              hipStream_t stream) {
    const float* x     = (const float*)d_in[0];   // (8,128,128,128) f32
    const float* xseqs = (const float*)d_in[1];   // (32,16384,128)  f32

    float* scanned = (float*)d_out;                               // 32*16384*128
    float* rec     = scanned + (size_t)NDIR * Nn * Ll * Cc;       // + 8*16384*128

    const int pixels = Nn * Ll;                                   // 131072
    dim3 blk(256);

    // scanned: each wave does NSLOT pixels -> 131072/(8*4) = 4096 blocks
    nss_scatter_async<<<pixels / (WAVES_PER_BLOCK * NSLOT), blk, 0, stream>>>(x, scanned);
    // rec: one wave per pixel -> 131072/8 = 16384 blocks
    nss_merge<<<pixels / WAVES_PER_BLOCK, blk, 0, stream>>>(xseqs, rec);
}